// Router_63900523430579
// MI455X (gfx1250) — compile-verified
//
#include <hip/hip_runtime.h>
#include <hip/hip_bf16.h>

typedef __attribute__((ext_vector_type(2))) float v2f;
typedef __attribute__((ext_vector_type(8))) float v8f;

#define DIM 2048
#define NEXP 64
#define TOK_PER_BLK 64

// One wave (32 lanes) per block; 64 tokens per block.
// GEMM: C[64 tok x 64 experts] via V_WMMA_F32_16X16X4_F32, 4x4 tile grid.
__global__ __launch_bounds__(32) void router_main_kernel(
    const float* __restrict__ x,     // [ntok, DIM]
    const float* __restrict__ W,     // [NEXP, DIM]
    float* __restrict__ out,         // [ntok*2 idx][ntok*2 w][1 aux]
    float* __restrict__ g_esum,      // [NEXP] global softmax-prob sums
    int ntok)
{
    __shared__ float sc[TOK_PER_BLK * 65];   // scores, stride 65 (bank-conflict free)
    __shared__ float esum_l[NEXP];

    const int lane = threadIdx.x;            // 0..31 (wave32)
    const long base_tok = (long)blockIdx.x * TOK_PER_BLK;

    esum_l[lane] = 0.0f;
    esum_l[lane + 32] = 0.0f;

    const int lm   = lane & 15;              // row within fragment
    const int koff = (lane >> 4) << 1;       // lanes 16..31 hold K={2,3}

    // Per-lane base pointers for A (tokens) and B (experts)
    const float* aptr[4];
    const float* bptr[4];
#pragma unroll
    for (int mi = 0; mi < 4; ++mi)
        aptr[mi] = x + (size_t)(base_tok + mi * 16 + lm) * DIM + koff;
#pragma unroll
    for (int ni = 0; ni < 4; ++ni)
        bptr[ni] = W + (size_t)(ni * 16 + lm) * DIM + koff;

    v8f acc[4][4];
#pragma unroll
    for (int mi = 0; mi < 4; ++mi)
#pragma unroll
        for (int ni = 0; ni < 4; ++ni)
            acc[mi][ni] = (v8f){0.f, 0.f, 0.f, 0.f, 0.f, 0.f, 0.f, 0.f};

    // K loop: 512 steps of K=4; 8 dword2 loads feed 16 WMMAs per step.
    for (int k = 0; k < DIM; k += 4) {
        v2f a[4], b[4];
#pragma unroll
        for (int ni = 0; ni < 4; ++ni)
            b[ni] = *(const v2f*)(bptr[ni] + k);
#pragma unroll
        for (int mi = 0; mi < 4; ++mi)
            a[mi] = *(const v2f*)(aptr[mi] + k);
#pragma unroll
        for (int mi = 0; mi < 4; ++mi) {
#pragma unroll
            for (int ni = 0; ni < 4; ++ni) {
                acc[mi][ni] = __builtin_amdgcn_wmma_f32_16x16x4_f32(
                    /*neg_a=*/false, a[mi],
                    /*neg_b=*/false, b[ni],
                    /*c_mod=*/(short)0, acc[mi][ni],
                    /*reuse_a=*/false, /*reuse_b=*/false);
            }
        }
    }

    // Spill C tiles to LDS.
    // C layout: VGPR v, lanes 0-15 -> M=v, lanes 16-31 -> M=v+8; N = lane&15.
    const int mhi = (lane >> 4) << 3;        // +8 rows for upper half-wave
#pragma unroll
    for (int mi = 0; mi < 4; ++mi) {
#pragma unroll
        for (int ni = 0; ni < 4; ++ni) {
#pragma unroll
            for (int v = 0; v < 8; ++v) {
                int row = mi * 16 + v + mhi;
                int col = ni * 16 + lm;
                sc[row * 65 + col] = acc[mi][ni][v];
            }
        }
    }
    __syncthreads();

    // Each lane finalizes 2 tokens: top-2 + softmax weights + full softmax for aux.
#pragma unroll
    for (int tt = 0; tt < 2; ++tt) {
        const int t = lane + tt * 32;
        const float* row = &sc[t * 65];

        float v1 = -3.402823e38f, v2 = -3.402823e38f;
        int i1 = 0, i2 = 0;
        for (int e = 0; e < NEXP; ++e) {
            float s = row[e];
            if (s > v1) { v2 = v1; i2 = i1; v1 = s; i1 = e; }
            else if (s > v2) { v2 = s; i2 = e; }
        }

        float ssum = 0.0f;
        for (int e = 0; e < NEXP; ++e)
            ssum += __expf(row[e] - v1);
        float inv = 1.0f / ssum;
        for (int e = 0; e < NEXP; ++e)
            atomicAdd(&esum_l[e], __expf(row[e] - v1) * inv);

        // softmax over the top-2 values (descending order)
        float e2 = __expf(v2 - v1);
        float w1 = 1.0f / (1.0f + e2);
        float w2 = 1.0f - w1;

        long gt = base_tok + t;
        long widx_base = (long)ntok * 2;
        out[gt * 2 + 0] = (float)i1;
        out[gt * 2 + 1] = (float)i2;
        out[widx_base + gt * 2 + 0] = w1;
        out[widx_base + gt * 2 + 1] = w2;
    }
    __syncthreads();

    atomicAdd(&g_esum[lane],      esum_l[lane]);
    atomicAdd(&g_esum[lane + 32], esum_l[lane + 32]);
}

__global__ __launch_bounds__(64) void router_aux_kernel(
    const float* __restrict__ g_esum, float* __restrict__ out_aux, int ntok)
{
    __shared__ float red[NEXP];
    int t = threadIdx.x;
    float usage = g_esum[t] * (1.0f / (float)ntok);
    red[t] = usage * usage;
    __syncthreads();
    for (int s = 32; s > 0; s >>= 1) {
        if (t < s) red[t] += red[t + s];
        __syncthreads();
    }
    if (t == 0) *out_aux = (float)NEXP * red[0];
}

extern "C" void kernel_launch(void* const* d_in, const int* in_sizes, int n_in,
                              void* d_out, int out_size, void* d_ws, size_t ws_size,
                              hipStream_t stream) {
    const float* x = (const float*)d_in[0];
    const float* W = (const float*)d_in[1];
    float* out = (float*)d_out;
    float* g_esum = (float*)d_ws;

    const int ntok = in_sizes[0] / DIM;              // 4*8192 = 32768
    const int nblk = ntok / TOK_PER_BLK;             // 512

    hipMemsetAsync(g_esum, 0, NEXP * sizeof(float), stream);
    router_main_kernel<<<nblk, 32, 0, stream>>>(x, W, out, g_esum, ntok);
    router_aux_kernel<<<1, 64, 0, stream>>>(g_esum, out + (out_size - 1), ntok);
}